// Softmax_3710851744412
// MI455X (gfx1250) — compile-verified
//
#include <hip/hip_runtime.h>
#include <math.h>

// ---------------------------------------------------------------------------
// Softmax over a flat fp32 vector, tuned for MI455X (gfx1250):
//  * memory-bound: ~268 MB unavoidable HBM traffic -> ~11.5 us floor
//    (x = 128 MiB fits the 192 MB L2, so pass 3's re-read hits L2)
//  * 4x b128 loads in flight per wave in the streaming loops (clause-able)
//  * wave32 shuffle reductions, NT store hints, global_prefetch_b8
//  * compile-time grid stride: no runtime integer division in the prologue
// ---------------------------------------------------------------------------

typedef __attribute__((ext_vector_type(4))) float v4f;

#define NBLOCKS 2048
#define TPB     256
constexpr int STRIDE = NBLOCKS * TPB;      // fixed launch geometry

// Merge two (max, sum-of-exp) online-softmax states; NaN-free for the
// (-inf, 0) identity element.
__device__ __forceinline__ void ms_combine(float& m, float& s, float m2, float s2) {
    float M = fmaxf(m, m2);
    if (M == -INFINITY) { s = s + s2; m = M; return; }
    s = s * __expf(m - M) + s2 * __expf(m2 - M);
    m = M;
}

// Wave32 + LDS block reduction of an (m, s) state. NWAVES = blockDim.x / 32.
template <int NWAVES>
__device__ __forceinline__ void block_reduce_ms(float& m, float& s) {
    __shared__ float sh_m[NWAVES];
    __shared__ float sh_s[NWAVES];
    const int lane = threadIdx.x & 31;
    const int wave = threadIdx.x >> 5;

#pragma unroll
    for (int off = 16; off > 0; off >>= 1) {
        float m2 = __shfl_xor(m, off, 32);
        float s2 = __shfl_xor(s, off, 32);
        ms_combine(m, s, m2, s2);
    }
    if (lane == 0) { sh_m[wave] = m; sh_s[wave] = s; }
    __syncthreads();
    if (wave == 0) {
        m = (lane < NWAVES) ? sh_m[lane] : -INFINITY;
        s = (lane < NWAVES) ? sh_s[lane] : 0.0f;
#pragma unroll
        for (int off = 16; off > 0; off >>= 1) {
            float m2 = __shfl_xor(m, off, 32);
            float s2 = __shfl_xor(s, off, 32);
            ms_combine(m, s, m2, s2);
        }
    }
}

__device__ __forceinline__ float max4(v4f v) {
    return fmaxf(fmaxf(v.x, v.y), fmaxf(v.z, v.w));
}
__device__ __forceinline__ float expsum4(v4f v, float M) {
    return (__expf(v.x - M) + __expf(v.y - M)) + (__expf(v.z - M) + __expf(v.w - M));
}

// ---------------------------------------------------------------------------
// Pass 1: streaming online-softmax partial reduction. Regular-temporal loads
// deliberately populate L2 so pass 3's re-read of x hits L2.
// ---------------------------------------------------------------------------
__global__ void __launch_bounds__(TPB)
softmax_partials(const float* __restrict__ x, float2* __restrict__ partials,
                 int n4, int rem) {
    const int tid = blockIdx.x * TPB + threadIdx.x;
    const v4f* __restrict__ x4 = (const v4f*)x;

    float m = -INFINITY;
    float s = 0.0f;

    const int iters   = n4 / STRIDE;       // compile-time shift, uniform
    const int batches = iters >> 2;        // 4 vectors (16 floats) per batch
    int i = tid;

    for (int b = 0; b < batches; ++b) {
        // 4 independent b128 loads in flight before any use (clause-able).
        v4f va = x4[i];
        v4f vb = x4[i +     STRIDE];
        v4f vc = x4[i + 2 * STRIDE];
        v4f vd = x4[i + 3 * STRIDE];
        // Prefetch the next batch's first line (gfx1250 global_prefetch_b8).
        if (b + 1 < batches)
            __builtin_prefetch((const void*)(x4 + i + 4 * STRIDE), 0, 0);

        float vm = fmaxf(fmaxf(max4(va), max4(vb)), fmaxf(max4(vc), max4(vd)));
        float M  = fmaxf(m, vm);
        // Single rescale per 16 elements; 16 independent v_exp_f32s.
        float t = (expsum4(va, M) + expsum4(vb, M)) +
                  (expsum4(vc, M) + expsum4(vd, M));
        s = s * __expf(m - M) + t;
        m = M;
        i += 4 * STRIDE;
    }
    // Leftover uniform iterations (iters % 4) + ragged end.
    for (; i < n4; i += STRIDE) {
        v4f v  = x4[i];
        float M = fmaxf(m, max4(v));
        s = s * __expf(m - M) + expsum4(v, M);
        m = M;
    }
    // Scalar tail (n % 4 != 0): first `rem` global threads.
    if (tid < rem) {
        float v = x[n4 * 4 + tid];
        float M = fmaxf(m, v);
        s = s * __expf(m - M) + __expf(v - M);
        m = M;
    }

    block_reduce_ms<TPB / 32>(m, s);
    if (threadIdx.x == 0) {
        float2 p; p.x = m; p.y = s;
        partials[blockIdx.x] = p;
    }
}

// ---------------------------------------------------------------------------
// Pass 2: single block folds all per-block partials into (M, 1/S).
// ---------------------------------------------------------------------------
__global__ void __launch_bounds__(1024)
softmax_finalize(const float2* __restrict__ partials, int nparts,
                 float2* __restrict__ out_ms) {
    float m = -INFINITY;
    float s = 0.0f;
    for (int i = threadIdx.x; i < nparts; i += 1024) {
        float2 p = partials[i];
        ms_combine(m, s, p.x, p.y);
    }
    block_reduce_ms<32>(m, s);
    if (threadIdx.x == 0) {
        float2 r; r.x = m; r.y = 1.0f / s;
        *out_ms = r;
    }
}

// ---------------------------------------------------------------------------
// Pass 3: out[i] = exp(x[i] - M) * invS. x re-read hits L2 (NT load: last
// use). Output stored non-temporally so 128 MiB of dirty result lines rinse
// to HBM instead of evicting x from L2 mid-pass.
// ---------------------------------------------------------------------------
__global__ void __launch_bounds__(TPB)
softmax_normalize(const float* __restrict__ x, float* __restrict__ out,
                  const float2* __restrict__ ms, int n4, int rem) {
    const int tid = blockIdx.x * TPB + threadIdx.x;
    const float2 p = *ms;                  // uniform -> scalar load + broadcast
    const float  M    = p.x;
    const float  invS = p.y;

    const v4f* __restrict__ x4 = (const v4f*)x;
    v4f* __restrict__ o4 = (v4f*)out;

    const int iters   = n4 / STRIDE;
    const int batches = iters >> 2;
    int i = tid;

    for (int b = 0; b < batches; ++b) {
        v4f a = __builtin_nontemporal_load(x4 + i);
        v4f c = __builtin_nontemporal_load(x4 + i +     STRIDE);
        v4f d = __builtin_nontemporal_load(x4 + i + 2 * STRIDE);
        v4f e = __builtin_nontemporal_load(x4 + i + 3 * STRIDE);
        if (b + 1 < batches)
            __builtin_prefetch((const void*)(x4 + i + 4 * STRIDE), 0, 0);

        v4f ra, rc, rd, re;
        ra.x = __expf(a.x - M) * invS; ra.y = __expf(a.y - M) * invS;
        ra.z = __expf(a.z - M) * invS; ra.w = __expf(a.w - M) * invS;
        rc.x = __expf(c.x - M) * invS; rc.y = __expf(c.y - M) * invS;
        rc.z = __expf(c.z - M) * invS; rc.w = __expf(c.w - M) * invS;
        rd.x = __expf(d.x - M) * invS; rd.y = __expf(d.y - M) * invS;
        rd.z = __expf(d.z - M) * invS; rd.w = __expf(d.w - M) * invS;
        re.x = __expf(e.x - M) * invS; re.y = __expf(e.y - M) * invS;
        re.z = __expf(e.z - M) * invS; re.w = __expf(e.w - M) * invS;

        __builtin_nontemporal_store(ra, o4 + i);
        __builtin_nontemporal_store(rc, o4 + i +     STRIDE);
        __builtin_nontemporal_store(rd, o4 + i + 2 * STRIDE);
        __builtin_nontemporal_store(re, o4 + i + 3 * STRIDE);
        i += 4 * STRIDE;
    }
    for (; i < n4; i += STRIDE) {
        v4f v = __builtin_nontemporal_load(x4 + i);
        v4f r;
        r.x = __expf(v.x - M) * invS; r.y = __expf(v.y - M) * invS;
        r.z = __expf(v.z - M) * invS; r.w = __expf(v.w - M) * invS;
        __builtin_nontemporal_store(r, o4 + i);
    }
    if (tid < rem) {
        int idx = n4 * 4 + tid;
        out[idx] = __expf(x[idx] - M) * invS;
    }
}

extern "C" void kernel_launch(void* const* d_in, const int* in_sizes, int n_in,
                              void* d_out, int out_size, void* d_ws, size_t ws_size,
                              hipStream_t stream) {
    const float* x   = (const float*)d_in[0];
    float*       out = (float*)d_out;
    const int n   = in_sizes[0];
    const int n4  = n >> 2;
    const int rem = n & 3;

    // Workspace layout: [0..255] final (M, 1/S); [256..] per-block partials.
    float2* final_ms = (float2*)d_ws;
    float2* partials = (float2*)((char*)d_ws + 256);

    softmax_partials <<<NBLOCKS, TPB,  0, stream>>>(x, partials, n4, rem);
    softmax_finalize <<<1,       1024, 0, stream>>>(partials, NBLOCKS, final_ms);
    softmax_normalize<<<NBLOCKS, TPB,  0, stream>>>(x, out, final_ms, n4, rem);
}